// MyLayer_90117003805128
// MI455X (gfx1250) — compile-verified
//
#include <hip/hip_runtime.h>

#define C_DIM 64
#define TWO_C 128
#define MID 256
#define OUT_DIM 64
#define BN_EPS 1e-5f

typedef __attribute__((ext_vector_type(2))) float v2f;
typedef __attribute__((ext_vector_type(8))) float v8f;

// D = A(16x4) * B(4x16) + C, full f32 precision (matches f32 reference).
__device__ __forceinline__ v8f wmma4(v2f a, v2f b, v8f c) {
  return __builtin_amdgcn_wmma_f32_16x16x4_f32(
      /*neg_a=*/false, a, /*neg_b=*/false, b,
      /*c_mod=*/(short)0, c, /*reuse_a=*/false, /*reuse_b=*/false);
}

// ---------------- Kernel 2: edge gather + scatter-add (L2-resident atomics) --
// 16 threads per edge; each handles a float4 chunk of the 64-wide feature.
__global__ void edge_scatter(const float* __restrict__ x,
                             const int* __restrict__ ei,   // [2, E]
                             const float* __restrict__ ea, // [E]
                             float* __restrict__ sums,     // [N, 128]
                             float* __restrict__ cnt,      // [N]
                             int E) {
  long long gid = (long long)blockIdx.x * blockDim.x + threadIdx.x;
  int e = (int)(gid >> 4);
  if (e >= E) return;
  int q = (int)(gid & 15);
  int row = ei[e];
  int col = ei[E + e];
  float w = ea[e];
  const float4 v = *reinterpret_cast<const float4*>(x + (size_t)col * C_DIM + q * 4);
  float* s = sums + (size_t)row * TWO_C + q * 4;
  atomicAdd(s + 0, v.x); atomicAdd(s + 1, v.y);
  atomicAdd(s + 2, v.z); atomicAdd(s + 3, v.w);
  float* s2 = s + C_DIM;
  atomicAdd(s2 + 0, v.x * w); atomicAdd(s2 + 1, v.y * w);
  atomicAdd(s2 + 2, v.z * w); atomicAdd(s2 + 3, v.w * w);
  if (q == 0) atomicAdd(cnt + row, 1.0f);
}

// ---------------- Kernel 3: agg = sums / max(cnt, 1) (in place) --------------
__global__ void finalize_agg(float* __restrict__ sums, const float* __restrict__ cnt,
                             long long total) {
  long long gid = (long long)blockIdx.x * blockDim.x + threadIdx.x;
  if (gid >= total) return;
  float c = cnt[gid >> 7];
  sums[gid] = sums[gid] / fmaxf(c, 1.0f);
}

// ---------------- Kernel 4: h = relu(agg @ W1 + b1) + BN column stats --------
// 16-row tile per block, 256 threads = 8 waves, each wave 2 16x16 col-tiles.
__global__ void __launch_bounds__(256)
gemm1_bn_stats(const float* __restrict__ agg, const float* __restrict__ W1,
               const float* __restrict__ b1, float* __restrict__ h,
               float* __restrict__ colsum, float* __restrict__ colsq, int N) {
  __shared__ float At[16][TWO_C + 4]; // +4 pad to spread LDS banks
  __shared__ float cs[MID];
  __shared__ float cq[MID];
  const int tid = threadIdx.x;
  const int rowBase = blockIdx.x * 16;
  for (int i = tid; i < 16 * TWO_C; i += 256) {
    int r = i >> 7, c = i & (TWO_C - 1);
    int row = rowBase + r;
    At[r][c] = (row < N) ? agg[(size_t)row * TWO_C + c] : 0.0f;
  }
  cs[tid] = 0.0f;
  cq[tid] = 0.0f;
  __syncthreads();

  const int wave = tid >> 5;
  const int lane = tid & 31;
  const int l16 = lane & 15;
  const int hi = lane >> 4; // 0: lanes carry K{0,1}; 1: lanes carry K{2,3}
  for (int t = 0; t < 2; ++t) {
    const int col0 = (wave * 2 + t) * 16;
    v8f acc = {};
    for (int kk = 0; kk < TWO_C / 4; ++kk) {
      const int kb = kk * 4 + hi * 2;
      v2f a, b;
      a.x = At[l16][kb];
      a.y = At[l16][kb + 1];
      b.x = W1[(size_t)kb * MID + col0 + l16];
      b.y = W1[(size_t)(kb + 1) * MID + col0 + l16];
      acc = wmma4(a, b, acc);
    }
    const int ncol = col0 + l16;
    const float bias = b1[ncol];
    for (int r = 0; r < 8; ++r) {
      int row = rowBase + r + hi * 8;
      float v = fmaxf(acc[r] + bias, 0.0f);
      if (row < N) {
        h[(size_t)row * MID + ncol] = v;
        atomicAdd(&cs[ncol], v);
        atomicAdd(&cq[ncol], v * v);
      }
    }
  }
  __syncthreads();
  atomicAdd(&colsum[tid], cs[tid]);
  atomicAdd(&colsq[tid], cq[tid]);
}

// ---------------- Kernel 5: fold batch-norm into GEMM2 weights ---------------
// s[c] = gamma[c]*rsqrt(var+eps); W2'[c][o] = s[c]*W2[c][o];
// b2'[o] = b2[o] + sum_c (beta[c] - mu[c]*s[c]) * W2[c][o]
__global__ void __launch_bounds__(256)
bn_prep(const float* __restrict__ colsum, const float* __restrict__ colsq,
        const float* __restrict__ gamma, const float* __restrict__ beta,
        const float* __restrict__ W2, const float* __restrict__ b2,
        float* __restrict__ W2p, float* __restrict__ b2p, int N) {
  __shared__ float b2acc[OUT_DIM];
  const int c = threadIdx.x;
  if (c < OUT_DIM) b2acc[c] = 0.0f;
  __syncthreads();
  const float invN = 1.0f / (float)N;
  const float mean = colsum[c] * invN;
  const float var = fmaxf(colsq[c] * invN - mean * mean, 0.0f);
  const float s = gamma[c] * rsqrtf(var + BN_EPS);
  const float t = beta[c] - mean * s;
  for (int o = 0; o < OUT_DIM; ++o) {
    float w = W2[c * OUT_DIM + o];
    W2p[c * OUT_DIM + o] = s * w;
    atomicAdd(&b2acc[o], t * w);
  }
  __syncthreads();
  if (c < OUT_DIM) b2p[c] = b2[c] + b2acc[c];
}

// ---------------- Kernel 6: out = h @ W2' + b2' ------------------------------
// 16-row tile per block, 128 threads = 4 waves, one 16x16 col-tile each.
__global__ void __launch_bounds__(128)
gemm2(const float* __restrict__ h, const float* __restrict__ W2p,
      const float* __restrict__ b2p, float* __restrict__ out, int N) {
  __shared__ float At[16][MID + 4];
  const int tid = threadIdx.x;
  const int rowBase = blockIdx.x * 16;
  for (int i = tid; i < 16 * MID; i += 128) {
    int r = i >> 8, c = i & (MID - 1);
    int row = rowBase + r;
    At[r][c] = (row < N) ? h[(size_t)row * MID + c] : 0.0f;
  }
  __syncthreads();

  const int wave = tid >> 5;
  const int lane = tid & 31;
  const int l16 = lane & 15;
  const int hi = lane >> 4;
  const int col0 = wave * 16;
  v8f acc = {};
  for (int kk = 0; kk < MID / 4; ++kk) {
    const int kb = kk * 4 + hi * 2;
    v2f a, b;
    a.x = At[l16][kb];
    a.y = At[l16][kb + 1];
    b.x = W2p[kb * OUT_DIM + col0 + l16];
    b.y = W2p[(kb + 1) * OUT_DIM + col0 + l16];
    acc = wmma4(a, b, acc);
  }
  const int ncol = col0 + l16;
  const float bias = b2p[ncol];
  for (int r = 0; r < 8; ++r) {
    int row = rowBase + r + hi * 8;
    if (row < N) out[(size_t)row * OUT_DIM + ncol] = acc[r] + bias;
  }
}

extern "C" void kernel_launch(void* const* d_in, const int* in_sizes, int n_in,
                              void* d_out, int out_size, void* d_ws, size_t ws_size,
                              hipStream_t stream) {
  const float* x     = (const float*)d_in[0];
  const int*   ei    = (const int*)d_in[1];
  const float* ea    = (const float*)d_in[2];
  // d_in[3] = batch (all zeros, unused by the reference math)
  const float* W1    = (const float*)d_in[4];
  const float* b1    = (const float*)d_in[5];
  const float* gamma = (const float*)d_in[6];
  const float* beta  = (const float*)d_in[7];
  const float* W2    = (const float*)d_in[8];
  const float* b2    = (const float*)d_in[9];
  float* out = (float*)d_out;

  const int N = in_sizes[0] / C_DIM;
  const int E = in_sizes[2];

  // Workspace carve-up (~154 MB): sums/agg, cnt, h, BN accumulators, W2', b2'
  char* ws = (char*)d_ws;
  size_t off = 0;
  float* sums   = (float*)(ws + off); off += (size_t)N * TWO_C * sizeof(float);
  float* cnt    = (float*)(ws + off); off += (size_t)N * sizeof(float);
  float* h      = (float*)(ws + off); off += (size_t)N * MID * sizeof(float);
  float* colsum = (float*)(ws + off); off += (size_t)MID * sizeof(float);
  float* colsq  = (float*)(ws + off); off += (size_t)MID * sizeof(float);
  float* W2p    = (float*)(ws + off); off += (size_t)MID * OUT_DIM * sizeof(float);
  float* b2p    = (float*)(ws + off); off += (size_t)OUT_DIM * sizeof(float);
  (void)off; (void)ws_size; (void)n_in; (void)out_size;

  // sums and cnt are contiguous -> one memset; BN accumulators -> one memset.
  hipMemsetAsync(sums, 0, (size_t)N * (TWO_C + 1) * sizeof(float), stream);
  hipMemsetAsync(colsum, 0, (size_t)2 * MID * sizeof(float), stream);

  {
    long long total = (long long)E * 16;
    int blocks = (int)((total + 255) / 256);
    edge_scatter<<<blocks, 256, 0, stream>>>(x, ei, ea, sums, cnt, E);
  }
  {
    long long total = (long long)N * TWO_C;
    int blocks = (int)((total + 255) / 256);
    finalize_agg<<<blocks, 256, 0, stream>>>(sums, cnt, total);
  }
  const int rowBlocks = (N + 15) / 16;
  gemm1_bn_stats<<<rowBlocks, 256, 0, stream>>>(sums, W1, b1, h, colsum, colsq, N);
  bn_prep<<<1, 256, 0, stream>>>(colsum, colsq, gamma, beta, W2, b2, W2p, b2p, N);
  gemm2<<<rowBlocks, 128, 0, stream>>>(h, W2p, b2p, out, N);
}